// ExactTisasTransformerBlock_3384434229568
// MI455X (gfx1250) — compile-verified
//
#include <hip/hip_runtime.h>

typedef __attribute__((ext_vector_type(2))) float v2f;
typedef __attribute__((ext_vector_type(4))) float v4f;
typedef __attribute__((ext_vector_type(8))) float v8f;

#define Bc  2
#define Tc  512
#define Hc  256
#define NHc 4
#define DKc 64

__constant__ constexpr float EPSc   = 1e-5f;
__constant__ constexpr float SCALEc = 0.125f;   // 1/sqrt(64)

// ---------------------------------------------------------------------------
// LayerNorm over H=256, one block (256 threads) per row.
// ---------------------------------------------------------------------------
__global__ void ln_kernel(const float* __restrict__ x, const float* __restrict__ g,
                          const float* __restrict__ bta, float* __restrict__ y) {
  __shared__ float red[256];
  const int row = blockIdx.x;
  const int t = threadIdx.x;
  const float v = x[(size_t)row * Hc + t];
  red[t] = v;
  __syncthreads();
  for (int s = 128; s > 0; s >>= 1) {
    if (t < s) red[t] += red[t + s];
    __syncthreads();
  }
  const float mu = red[0] * (1.0f / Hc);
  __syncthreads();
  const float d = v - mu;
  red[t] = d * d;
  __syncthreads();
  for (int s = 128; s > 0; s >>= 1) {
    if (t < s) red[t] += red[t + s];
    __syncthreads();
  }
  const float var = red[0] * (1.0f / Hc);
  y[(size_t)row * Hc + t] = d * rsqrtf(var + EPSc) * g[t] + bta[t];
}

// ---------------------------------------------------------------------------
// Generic WMMA f32 GEMM-NT:  C = alpha * (A · B^T) + bias[col] (opt ReLU)
//                                + extra[row,col]
// One wave (32 threads) per 16x16 tile. Batched via blockIdx.y/z strides.
// A: [M,K] lda   B: [N,K] ldb   C/extra: [M,N] ldc
// Uses V_WMMA_F32_16X16X4_F32 (native fp32 matrix op, CDNA5).
// ---------------------------------------------------------------------------
__global__ void wmma_gemm_nt(const float* __restrict__ A, const float* __restrict__ Bm,
                             const float* __restrict__ bias, const float* __restrict__ extra,
                             float* __restrict__ C,
                             int M, int N, int K, int lda, int ldb, int ldc,
                             long long sAy, long long sAz,
                             long long sBy, long long sBz,
                             long long sCy, long long sCz,
                             float alpha, int relu) {
  const int lane   = threadIdx.x;            // 32 threads = 1 wave, EXEC all ones
  const int ntiles = N >> 4;
  const int tn = (blockIdx.x % ntiles) << 4;
  const int tm = (blockIdx.x / ntiles) << 4;

  const long long offC = (long long)blockIdx.y * sCy + (long long)blockIdx.z * sCz;
  const float* Ab = A  + (long long)blockIdx.y * sAy + (long long)blockIdx.z * sAz;
  const float* Bb = Bm + (long long)blockIdx.y * sBy + (long long)blockIdx.z * sBz;
  float*       Cb = C + offC;

  const int r     = lane & 15;          // M (for A) / N (for B) within tile
  const int khalf = (lane >> 4) << 1;   // lanes 16-31 hold K=k+2,k+3

  const float* ap = Ab + (size_t)(tm + r) * lda + khalf;
  const float* bp = Bb + (size_t)(tn + r) * ldb + khalf;

  v8f acc = {};
  for (int k = 0; k < K; k += 4) {
    const v2f a = *(const v2f*)(ap + k);
    const v2f b = *(const v2f*)(bp + k);
    acc = __builtin_amdgcn_wmma_f32_16x16x4_f32(
        /*neg_a=*/false, a, /*neg_b=*/false, b,
        /*c_mod=*/(short)0, acc, /*reuse_a=*/false, /*reuse_b=*/false);
  }

  const int col = tn + r;
  const float bi = bias ? bias[col] : 0.0f;
  const int rowbase = tm + ((lane >> 4) << 3);   // lanes 16-31 hold M=8..15
#pragma unroll
  for (int rr = 0; rr < 8; ++rr) {
    const int row = rowbase + rr;
    float t = alpha * acc[rr] + bi;
    if (relu) t = fmaxf(t, 0.0f);
    if (extra) t += extra[offC + (size_t)row * ldc + col];
    Cb[(size_t)row * ldc + col] = t;
  }
}

// ---------------------------------------------------------------------------
// Streaming attention kernel: one block (256 threads, 8 waves) per (b, i).
// Phase 1: s[n][j] = sbase[b,n,i,j] + SCALE * q_i(n) . rk[b,i,j,n]  (j <= i)
//          each wave owns one j; lane l covers channels l*8..l*8+7 (one 1KB
//          contiguous line of r_k per j -> fully coalesced stream).
// Phase 2: softmax per head (wave-level shfl reductions, wave32).
// Phase 3: out[h] = sum_j p[n][j]*(vp[b,j,h] + rv[b,i,j,h]); x2 = qn + out.
// r_k / r_v are single-use streams (537 MB causal half) -> non-temporal loads
// (TH=NT) so they don't evict the L2-resident vp / sbase / Qp working set.
// ---------------------------------------------------------------------------
__global__ void attn_kernel(const float* __restrict__ Qp, const float* __restrict__ vp,
                            const float* __restrict__ rk, const float* __restrict__ rv,
                            const float* __restrict__ sbase, const float* __restrict__ qn,
                            float* __restrict__ x2) {
  __shared__ float ps[NHc][Tc];
  const int b = blockIdx.x / Tc;
  const int i = blockIdx.x % Tc;
  const int tid = threadIdx.x;
  const int w = tid >> 5;
  const int l = tid & 31;

  const float* qrow = Qp + ((size_t)b * Tc + i) * Hc;
  const v4f q0 = *(const v4f*)(qrow + l * 8);
  const v4f q1 = *(const v4f*)(qrow + l * 8 + 4);

  const float* rkrow = rk + ((size_t)b * Tc + i) * Tc * Hc;
  const float* sb = sbase + (size_t)b * NHc * Tc * Tc + (size_t)i * Tc;

  for (int jb = 0; jb <= i; jb += 8) {
    const int j = jb + w;
    if (j <= i) {
      const float* rp = rkrow + (size_t)j * Hc + l * 8;
      const v4f r0 = __builtin_nontemporal_load((const v4f*)(rp));
      const v4f r1 = __builtin_nontemporal_load((const v4f*)(rp + 4));
      float part = q0.x * r0.x + q0.y * r0.y + q0.z * r0.z + q0.w * r0.w
                 + q1.x * r1.x + q1.y * r1.y + q1.z * r1.z + q1.w * r1.w;
      part += __shfl_xor(part, 1, 32);
      part += __shfl_xor(part, 2, 32);
      part += __shfl_xor(part, 4, 32);           // 8-lane group = one head
      if ((l & 7) == 0) {
        const int n = l >> 3;
        ps[n][j] = sb[(size_t)n * Tc * Tc + j] + SCALEc * part;
      }
    }
  }
  __syncthreads();

  if (w < NHc) {                                  // one wave per head
    const int n = w;
    float mx = -1e30f;
    for (int j = l; j <= i; j += 32) mx = fmaxf(mx, ps[n][j]);
    for (int off = 16; off; off >>= 1) mx = fmaxf(mx, __shfl_xor(mx, off, 32));
    float sum = 0.0f;
    for (int j = l; j <= i; j += 32) {
      const float e = __expf(ps[n][j] - mx);
      ps[n][j] = e;
      sum += e;
    }
    for (int off = 16; off; off >>= 1) sum += __shfl_xor(sum, off, 32);
    const float inv = 1.0f / sum;
    for (int j = l; j <= i; j += 32) ps[n][j] *= inv;
  }
  __syncthreads();

  const int h = tid;
  const int n = h >> 6;
  const float* rvc = rv + ((size_t)b * Tc + i) * Tc * Hc + h;
  const float* vpc = vp + (size_t)b * Tc * Hc + h;
  float acc = 0.0f;
  int j = 0;
  for (; j + 4 <= i + 1; j += 4) {
    const float p0 = ps[n][j], p1 = ps[n][j + 1], p2 = ps[n][j + 2], p3 = ps[n][j + 3];
    const float r0 = __builtin_nontemporal_load(rvc + (size_t)(j    ) * Hc);
    const float r1 = __builtin_nontemporal_load(rvc + (size_t)(j + 1) * Hc);
    const float r2 = __builtin_nontemporal_load(rvc + (size_t)(j + 2) * Hc);
    const float r3 = __builtin_nontemporal_load(rvc + (size_t)(j + 3) * Hc);
    const float a0 = vpc[(size_t)(j    ) * Hc] + r0;
    const float a1 = vpc[(size_t)(j + 1) * Hc] + r1;
    const float a2 = vpc[(size_t)(j + 2) * Hc] + r2;
    const float a3 = vpc[(size_t)(j + 3) * Hc] + r3;
    acc = fmaf(p0, a0, fmaf(p1, a1, fmaf(p2, a2, fmaf(p3, a3, acc))));
  }
  for (; j <= i; ++j)
    acc = fmaf(ps[n][j],
               vpc[(size_t)j * Hc] + __builtin_nontemporal_load(rvc + (size_t)j * Hc),
               acc);

  const size_t oidx = ((size_t)b * Tc + i) * Hc + h;
  x2[oidx] = qn[oidx] + acc;                      // residual uses normed q
}

// ---------------------------------------------------------------------------
extern "C" void kernel_launch(void* const* d_in, const int* in_sizes, int n_in,
                              void* d_out, int out_size, void* d_ws, size_t ws_size,
                              hipStream_t stream) {
  (void)in_sizes; (void)n_in; (void)out_size; (void)ws_size;
  const float* x     = (const float*)d_in[0];
  const float* pos_k = (const float*)d_in[1];
  const float* pos_v = (const float*)d_in[2];
  const float* r_k   = (const float*)d_in[3];
  const float* r_v   = (const float*)d_in[4];
  // d_in[5] = mask (causal, handled analytically)
  const float* Wq  = (const float*)d_in[6];
  const float* bq  = (const float*)d_in[7];
  const float* Wk  = (const float*)d_in[8];
  const float* bk  = (const float*)d_in[9];
  const float* Wv  = (const float*)d_in[10];
  const float* bv  = (const float*)d_in[11];
  const float* g1  = (const float*)d_in[12];
  const float* b1  = (const float*)d_in[13];
  const float* g2  = (const float*)d_in[14];
  const float* b2  = (const float*)d_in[15];
  const float* W1  = (const float*)d_in[16];
  const float* bf1 = (const float*)d_in[17];
  const float* W2  = (const float*)d_in[18];
  const float* bf2 = (const float*)d_in[19];

  const size_t ROW = (size_t)Bc * Tc * Hc;     // 262144 floats
  float* ws   = (float*)d_ws;
  float* qn    = ws;
  float* Qp    = ws + 1 * ROW;
  float* kp    = ws + 2 * ROW;
  float* vp    = ws + 3 * ROW;
  float* x2    = ws + 4 * ROW;
  float* z     = ws + 5 * ROW;
  float* h1    = ws + 6 * ROW;
  float* sbase = ws + 7 * ROW;                 // B*NH*T*T = 2,097,152 floats

  const int M = Bc * Tc;                       // 1024

  // 1) q = LN1(x)
  ln_kernel<<<Bc * Tc, 256, 0, stream>>>(x, g1, b1, qn);

  // 2) projections (kp/vp fold pos_k/pos_v via 'extra')
  dim3 gp((M / 16) * (Hc / 16), 1, 1);
  wmma_gemm_nt<<<gp, 32, 0, stream>>>(qn, Wq, bq, nullptr, Qp,
      M, Hc, Hc, Hc, Hc, Hc, 0, 0, 0, 0, 0, 0, 1.0f, 0);
  wmma_gemm_nt<<<gp, 32, 0, stream>>>(x, Wk, bk, pos_k, kp,
      M, Hc, Hc, Hc, Hc, Hc, 0, 0, 0, 0, 0, 0, 1.0f, 0);
  wmma_gemm_nt<<<gp, 32, 0, stream>>>(x, Wv, bv, pos_v, vp,
      M, Hc, Hc, Hc, Hc, Hc, 0, 0, 0, 0, 0, 0, 1.0f, 0);

  // 3) sbase[b,n,i,j] = SCALE * Q_i(n) . kp_j(n)   (batched over b=gridY, n=gridZ)
  dim3 gs((Tc / 16) * (Tc / 16), Bc, NHc);
  wmma_gemm_nt<<<gs, 32, 0, stream>>>(Qp, kp, nullptr, nullptr, sbase,
      Tc, Tc, DKc, Hc, Hc, Tc,
      (long long)Tc * Hc, DKc,
      (long long)Tc * Hc, DKc,
      (long long)NHc * Tc * Tc, (long long)Tc * Tc,
      SCALEc, 0);

  // 4) streaming attention + residual -> x2
  attn_kernel<<<Bc * Tc, 256, 0, stream>>>(Qp, vp, r_k, r_v, sbase, qn, x2);

  // 5) z = LN2(x2)
  ln_kernel<<<Bc * Tc, 256, 0, stream>>>(x2, g2, b2, z);

  // 6) FFN: h1 = relu(z@W1^T + bf1);  out = z + h1@W2^T + bf2
  wmma_gemm_nt<<<gp, 32, 0, stream>>>(z, W1, bf1, nullptr, h1,
      M, Hc, Hc, Hc, Hc, Hc, 0, 0, 0, 0, 0, 0, 1.0f, 1);
  wmma_gemm_nt<<<gp, 32, 0, stream>>>(h1, W2, bf2, z, (float*)d_out,
      M, Hc, Hc, Hc, Hc, Hc, 0, 0, 0, 0, 0, 0, 1.0f, 0);
}